// PerElementSelfInteraction_8847632630531
// MI455X (gfx1250) — compile-verified
//
#include <hip/hip_runtime.h>

typedef float v2f __attribute__((ext_vector_type(2)));
typedef float v8f __attribute__((ext_vector_type(8)));
typedef int   v4i __attribute__((ext_vector_type(4)));

#define N_ATOMS_C   100000
#define N_ELEM      10
#define DIM         480
#define TILE_ATOMS  32
#define XS_STRIDE   484          // even, 16B-aligned rows; 484 mod 64 = 36, gcd(36,64)=4 -> 16 rows hit 16 distinct banks
#define MAX_TILES   3200         // >= 100000/32 + 10

// workspace layout in 4-byte units
#define WS_COUNTS   0
#define WS_CURSOR   16
#define WS_NUMTILES 32
#define WS_TSPEC    64
#define WS_TBASE    (WS_TSPEC + MAX_TILES)             // 3264
#define WS_TLEN     (WS_TBASE + MAX_TILES)             // 6464
#define WS_PERM     (WS_TLEN + MAX_TILES)              // 9664
#define WS_WT0      (WS_PERM + N_ATOMS_C + 32)        // 109696 (16B aligned)
#define WS_WT1      (WS_WT0 + 10*128*128)              // +163840
#define WS_WT2      (WS_WT1 + 10*64*64)                // +40960

#if defined(__has_builtin)
#if __has_builtin(__builtin_amdgcn_global_load_async_to_lds_b128)
#define HAVE_ASYNC_LDS 1
#endif
#if __has_builtin(__builtin_amdgcn_s_wait_asynccnt)
#define HAVE_WAIT_ASYNC 1
#endif
#endif

// ---------------- histogram (LDS-aggregated: <=10 global atomics per block) ----------------
__global__ void k_hist(const int* __restrict__ an, int* __restrict__ counts) {
    __shared__ int lc[N_ELEM];
    if (threadIdx.x < N_ELEM) lc[threadIdx.x] = 0;
    __syncthreads();
    int i = blockIdx.x * blockDim.x + threadIdx.x;
    if (i < N_ATOMS_C) atomicAdd(&lc[an[i]], 1);
    __syncthreads();
    if (threadIdx.x < N_ELEM && lc[threadIdx.x] != 0)
        atomicAdd(&counts[threadIdx.x], lc[threadIdx.x]);
}

// ---------------- weight transpose: WT[e][v][u] = W[e][u][v] ----------------
__global__ void k_transpose(const float* __restrict__ W0, const float* __restrict__ W1,
                            const float* __restrict__ W2, float* __restrict__ wt0,
                            float* __restrict__ wt1, float* __restrict__ wt2) {
    int i = blockIdx.x * 256 + threadIdx.x;
    if (i < 163840) {                       // 10 x 128 x 128
        int e = i >> 14, r = i & 16383, v = r >> 7, u = r & 127;
        wt0[i] = W0[(e << 14) + (u << 7) + v];
    } else if (i < 204800) {                // 10 x 64 x 64
        int j = i - 163840;
        int e = j >> 12, r = j & 4095, v = r >> 6, u = r & 63;
        wt1[j] = W1[(e << 12) + (u << 6) + v];
    } else if (i < 215040) {                // 10 x 32 x 32
        int j = i - 204800;
        int e = j >> 10, r = j & 1023, v = r >> 5, u = r & 31;
        wt2[j] = W2[(e << 10) + (u << 5) + v];
    }
}

// ---------------- prefix + tile plan (parallel fill) ----------------
__global__ void k_plan(int* __restrict__ ws) {
    __shared__ int sOff[N_ELEM + 1];   // atom offsets per species
    __shared__ int sTOff[N_ELEM + 1];  // tile offsets per species
    const int tid = threadIdx.x;
    if (tid == 0) {
        int off = 0, toff = 0;
        for (int e = 0; e < N_ELEM; ++e) {
            sOff[e] = off; sTOff[e] = toff;
            int c = ws[WS_COUNTS + e];
            ws[WS_CURSOR + e] = off;
            off += c;
            toff += (c + TILE_ATOMS - 1) / TILE_ATOMS;
        }
        sOff[N_ELEM] = off; sTOff[N_ELEM] = toff;
        ws[WS_NUMTILES] = toff;
    }
    __syncthreads();
    const int nt = sTOff[N_ELEM];
    for (int t = tid; t < nt; t += blockDim.x) {
        int e = 0;
        while (t >= sTOff[e + 1]) ++e;
        int base = sOff[e] + (t - sTOff[e]) * TILE_ATOMS;
        int rem = sOff[e + 1] - base;
        ws[WS_TSPEC + t] = e;
        ws[WS_TBASE + t] = base;
        ws[WS_TLEN + t] = (rem < TILE_ATOMS) ? rem : TILE_ATOMS;
    }
}

// ---------------- scatter (two-level: LDS ranks, <=10 global atomics per block) ----------------
__global__ void k_scatter(const int* __restrict__ an, int* __restrict__ ws) {
    __shared__ int lc[N_ELEM];
    __shared__ int lbase[N_ELEM];
    if (threadIdx.x < N_ELEM) lc[threadIdx.x] = 0;
    __syncthreads();
    int i = blockIdx.x * blockDim.x + threadIdx.x;
    int e = 0, rank = 0;
    bool valid = (i < N_ATOMS_C);
    if (valid) { e = an[i]; rank = atomicAdd(&lc[e], 1); }
    __syncthreads();
    if (threadIdx.x < N_ELEM && lc[threadIdx.x] != 0)
        lbase[threadIdx.x] = atomicAdd(&ws[WS_CURSOR + threadIdx.x], lc[threadIdx.x]);
    __syncthreads();
    if (valid) ws[WS_PERM + lbase[e] + rank] = i;
}

// ---------------- one 16x16 output tile of W^T (MxK) * X (KxN) via f32 WMMA ----------------
// A (16x4): lane L: M=L&15; VGPR0/1 hold K=kbase, kbase+1, kbase=(L>>4)*2
// B (4x16): lane L: N=L&15; VGPR0/1 hold K=kbase, kbase+1
// C/D (16x16): VGPR j: lanes 0-15 -> M=j, lanes 16-31 -> M=j+8; N=L&15
template<int MUL, int D, int OFF>
__device__ __forceinline__ void do_tile(const float* __restrict__ wtE,
                                        const float* __restrict__ xs,
                                        const int* __restrict__ aid, int len,
                                        int mtile, int ntile, int lane,
                                        float norm, float* __restrict__ out) {
    const int mrow  = lane & 15;
    const int kbase = (lane >> 4) << 1;
    const int nIdx  = ntile * 16 + (lane & 15);
    const int aLoc  = nIdx / D;
    const int m     = nIdx % D;
    const float* xcol = xs + aLoc * XS_STRIDE + OFF + m;
    const float* arow = wtE + (mtile * 16 + mrow) * MUL + kbase;

    v8f acc = {};
#pragma unroll
    for (int kk = 0; kk < MUL / 4; ++kk) {
        v2f a = *(const v2f*)(arow + kk * 4);            // global_load_b64 (L2-resident W)
        v2f b;
        b.x = xcol[(kk * 4 + kbase + 0) * D];            // ds_load_2addr_b32 (fused)
        b.y = xcol[(kk * 4 + kbase + 1) * D];
        acc = __builtin_amdgcn_wmma_f32_16x16x4_f32(
            false, a, false, b, (short)0, acc, false, false);
    }
    if (aLoc < len) {
        const int atom = aid[aLoc];
        float* orow = out + atom * DIM + OFF + m;
        const int mbase = mtile * 16 + ((lane >> 4) << 3);
#pragma unroll
        for (int j = 0; j < 8; ++j)
            orow[(mbase + j) * D] = acc[j] * norm;
    }
}

// ---------------- grouped GEMM: one block = 32 same-species atoms, 8 waves ----------------
__global__ void __launch_bounds__(256) k_gemm(const float* __restrict__ emb,
                                              const int* __restrict__ ws,
                                              const float* __restrict__ wt0,
                                              const float* __restrict__ wt1,
                                              const float* __restrict__ wt2,
                                              float* __restrict__ out) {
    __shared__ float xs[TILE_ATOMS * XS_STRIDE];         // ~60.5 KB
    __shared__ int   aid[TILE_ATOMS];

    const int t = blockIdx.x;
    if (t >= ws[WS_NUMTILES]) return;                    // block-uniform
    const int e    = ws[WS_TSPEC + t];
    const int base = ws[WS_TBASE + t];
    const int len  = ws[WS_TLEN  + t];
    const int tid  = threadIdx.x;

    if (tid < TILE_ATOMS) {
        int idx = (tid < len) ? tid : 0;                 // pad with a valid atom
        aid[tid] = ws[WS_PERM + base + idx];
    }
    __syncthreads();

    const float* w0e = wt0 + e * (128 * 128);
    const float* w1e = wt1 + e * (64 * 64);
    const float* w2e = wt2 + e * (32 * 32);
    __builtin_prefetch(w0e, 0, 0);                       // global_prefetch into caches

    // stage X tile HBM -> LDS
#if defined(HAVE_ASYNC_LDS)
    for (int i = tid; i < TILE_ATOMS * (DIM / 4); i += 256) {
        int a = i / (DIM / 4), q = i - a * (DIM / 4);    // q in 16-byte units
        const float* g = emb + (size_t)aid[a] * DIM + q * 4;
        float* l = xs + a * XS_STRIDE + q * 4;           // 16B-aligned (stride 484)
        __builtin_amdgcn_global_load_async_to_lds_b128(
            (v4i*)g, (v4i*)l, 0, 0);
    }
#if defined(HAVE_WAIT_ASYNC)
    __builtin_amdgcn_s_wait_asynccnt(0);
#else
    asm volatile("s_wait_asynccnt 0x0" ::: "memory");
#endif
#else
    for (int i = tid; i < TILE_ATOMS * DIM; i += 256) {
        int a = i / DIM, c = i - a * DIM;
        xs[a * XS_STRIDE + c] = emb[(size_t)aid[a] * DIM + c];
    }
#endif
    __syncthreads();

    const int wave = tid >> 5;
    const int lane = tid & 31;
    // 60 output tiles: irrep0 8Mx2N=16, irrep1 4Mx6N=24, irrep2 2Mx10N=20
    for (int tt = wave; tt < 60; tt += 8) {              // wave-uniform loop
        if (tt < 16) {
            do_tile<128, 1, 0>(w0e, xs, aid, len, tt >> 1, tt & 1, lane, 0.0279508497f, out);
        } else if (tt < 40) {
            int j = tt - 16;
            do_tile<64, 3, 128>(w1e, xs, aid, len, j / 6, j % 6, lane, 0.0395284707f, out);
        } else {
            int j = tt - 40;
            do_tile<32, 5, 320>(w2e, xs, aid, len, j / 10, j % 10, lane, 0.0559016994f, out);
        }
    }
}

extern "C" void kernel_launch(void* const* d_in, const int* in_sizes, int n_in,
                              void* d_out, int out_size, void* d_ws, size_t ws_size,
                              hipStream_t stream) {
    const float* emb = (const float*)d_in[0];
    const int*   an  = (const int*)d_in[1];
    const float* W0  = (const float*)d_in[2];
    const float* W1  = (const float*)d_in[3];
    const float* W2  = (const float*)d_in[4];
    float* out = (float*)d_out;
    int*   ws  = (int*)d_ws;
    float* wsF = (float*)d_ws;

    (void)hipMemsetAsync(d_ws, 0, 64 * sizeof(int), stream);  // counts+cursor+numTiles

    k_hist<<<(N_ATOMS_C + 255) / 256, 256, 0, stream>>>(an, ws + WS_COUNTS);
    k_transpose<<<(215040 + 255) / 256, 256, 0, stream>>>(
        W0, W1, W2, wsF + WS_WT0, wsF + WS_WT1, wsF + WS_WT2);
    k_plan<<<1, 256, 0, stream>>>(ws);
    k_scatter<<<(N_ATOMS_C + 255) / 256, 256, 0, stream>>>(an, ws);
    k_gemm<<<3135, 256, 0, stream>>>(emb, ws, wsF + WS_WT0, wsF + WS_WT1,
                                     wsF + WS_WT2, out);
}